// MolecularIGCN_53068615909527
// MI455X (gfx1250) — compile-verified
//
#include <hip/hip_runtime.h>
#include <hip/hip_bf16.h>

#define N_NODES 32768
#define N_EDGES 262144
#define F_IN    74
#define D0      128
#define D1      256
#define BN_EPS  1e-5f

typedef __attribute__((ext_vector_type(2))) float v2f;
typedef __attribute__((ext_vector_type(8))) float v8f;

__device__ __forceinline__ v8f wmma_f32(v2f a, v2f b, v8f c) {
    // V_WMMA_F32_16X16X4_F32: D = A(16x4) * B(4x16) + C(16x16), all fp32
    return __builtin_amdgcn_wmma_f32_16x16x4_f32(
        false, a, false, b, (short)0, c, false, false);
}

__device__ __forceinline__ v2f ld2(const float* p) {
    v2f r; r.x = p[0]; r.y = p[1]; return r;
}

__device__ __forceinline__ float gelu_exact(float x) {
    return 0.5f * x * (1.0f + erff(x * 0.7071067811865476f));
}

// ---------------- utility kernels ----------------

__global__ __launch_bounds__(256) void k_zero(float* __restrict__ p, int n) {
    int i = blockIdx.x * 256 + threadIdx.x;
    if (i < n) p[i] = 0.0f;
}

__global__ __launch_bounds__(256) void k_deg(const int* __restrict__ src,
                                             const int* __restrict__ dst,
                                             float* __restrict__ deg_out,
                                             float* __restrict__ deg_in) {
    int e = blockIdx.x * 256 + threadIdx.x;
    if (e < N_EDGES) {
        atomicAdd(&deg_out[src[e]], 1.0f);
        atomicAdd(&deg_in[dst[e]], 1.0f);
    }
}

__global__ __launch_bounds__(256) void k_dinv(const float* __restrict__ deg,
                                              float* __restrict__ dinv, int n) {
    int i = blockIdx.x * 256 + threadIdx.x;
    if (i < n) {
        float d = deg[i];
        dinv[i] = rsqrtf(d < 1.0f ? 1.0f : d);
    }
}

// scatter: agg[dst] += x[src] * dinv_out[src], vectorized over float4
template<int D>
__global__ __launch_bounds__(256) void k_scatter(const float* __restrict__ x,
                                                 const float* __restrict__ dinv_out,
                                                 const int* __restrict__ src,
                                                 const int* __restrict__ dst,
                                                 float* __restrict__ agg) {
    constexpr int V = D / 4;
    int idx = blockIdx.x * 256 + threadIdx.x;
    if (idx >= N_EDGES * V) return;
    int e = idx / V;            // V is a power of two -> shift
    int f = (idx - e * V) * 4;
    int s = src[e], t = dst[e];
    float sc = dinv_out[s];
    const float4 v = *(const float4*)(x + s * D + f);
    float* p = agg + t * D + f;
    atomicAdd(p + 0, v.x * sc);
    atomicAdd(p + 1, v.y * sc);
    atomicAdd(p + 2, v.z * sc);
    atomicAdd(p + 3, v.w * sc);
}

// per-column sums / sum-of-squares over a [N_NODES, 256] tensor
__global__ __launch_bounds__(256) void k_colstats(const float* __restrict__ t,
                                                  float* __restrict__ sum,
                                                  float* __restrict__ sumsq) {
    int col  = threadIdx.x;         // 256 threads = 256 cols (coalesced)
    int row0 = blockIdx.x * 128;    // 256 blocks * 128 rows
    float s = 0.0f, q = 0.0f;
    for (int r = 0; r < 128; ++r) {
        float v = t[(row0 + r) * D1 + col];
        s += v;
        q += v * v;
    }
    atomicAdd(&sum[col], s);
    atomicAdd(&sumsq[col], q);
}

__global__ __launch_bounds__(256) void k_bn(const float* __restrict__ t,
                                            const float* __restrict__ sum,
                                            const float* __restrict__ sumsq,
                                            const float* __restrict__ g,
                                            const float* __restrict__ b,
                                            float* __restrict__ out) {
    int i = blockIdx.x * 256 + threadIdx.x;
    if (i >= N_NODES * D1) return;
    int c = i & (D1 - 1);
    const float inv_n = 1.0f / (float)N_NODES;
    float m = sum[c] * inv_n;
    float v = sumsq[c] * inv_n - m * m;
    out[i] = (t[i] - m) * rsqrtf(v + BN_EPS) * g[c] + b[c];
}

// ---------------- WMMA GEMM kernels ----------------

// h0 = node_feats @ W_init.T   [N,74] x [74,128] -> [N,128], K=74 tail-masked
__global__ __launch_bounds__(256) void k_init_gemm(const float* __restrict__ nf,
                                                   const float* __restrict__ Wi,
                                                   float* __restrict__ h0) {
    const int lane = threadIdx.x & 31;
    const int wave = blockIdx.x * 8 + (threadIdx.x >> 5); // 4096 waves
    const int tm = wave >> 1;        // 0..2047 (16-row tiles)
    const int tn = wave & 1;         // 0..1   (64-col tiles)
    const int m0 = tm * 16, n0 = tn * 64;
    const int half = lane >> 4, l = lane & 15;

    v8f acc[4] = {};
    const float* pa = nf + (m0 + l) * F_IN;

    for (int k0 = 0; k0 < F_IN; k0 += 4) {
        const int ka = k0 + 2 * half;
        const bool ok = (ka + 1) < F_IN;       // per-lane tail mask (cndmask, EXEC intact)
        v2f a;
        a.x = ok ? pa[ka]     : 0.0f;
        a.y = ok ? pa[ka + 1] : 0.0f;
#pragma unroll
        for (int nt = 0; nt < 4; ++nt) {
            const int col = n0 + nt * 16 + l;
            v2f b;
            b.x = ok ? Wi[col * F_IN + ka]     : 0.0f;
            b.y = ok ? Wi[col * F_IN + ka + 1] : 0.0f;
            acc[nt] = wmma_f32(a, b, acc[nt]);
        }
    }
#pragma unroll
    for (int nt = 0; nt < 4; ++nt) {
        const int col = n0 + nt * 16 + l;
#pragma unroll
        for (int r = 0; r < 8; ++r) {
            const int row = m0 + r + 8 * half;
            h0[row * D0 + col] = acc[nt][r];
        }
    }
}

// layer1 pre-BN: t1 = gelu((agg1*dinv_in) @ w1 + b1) + (h0 @ res_w1.T + res_b1)
__global__ __launch_bounds__(256) void k_layer1(const float* __restrict__ h0,
                                                const float* __restrict__ agg1,
                                                const float* __restrict__ dinv_in,
                                                const float* __restrict__ w1,   // [128,256]
                                                const float* __restrict__ b1,   // [256]
                                                const float* __restrict__ rw,   // [256,128]
                                                const float* __restrict__ rb,   // [256]
                                                float* __restrict__ t1) {
    const int lane = threadIdx.x & 31;
    const int wave = blockIdx.x * 8 + (threadIdx.x >> 5); // 8192 waves
    const int tm = wave >> 2;        // 0..2047
    const int tn = wave & 3;         // 0..3
    const int m0 = tm * 16, n0 = tn * 64;
    const int half = lane >> 4, l = lane & 15;

    const float di = dinv_in[m0 + l];   // A-row norm, constant over k-loop

    v8f accg[4] = {};
    v8f accr[4] = {};
    const float* pa1 = agg1 + (m0 + l) * D0 + 2 * half;
    const float* pa2 = h0   + (m0 + l) * D0 + 2 * half;

    for (int k0 = 0; k0 < D0; k0 += 4) {
        v2f a1 = ld2(pa1 + k0);
        a1.x *= di; a1.y *= di;
        v2f a2 = ld2(pa2 + k0);
        const int kb = k0 + 2 * half;
        const float* pb = w1 + kb * D1;
#pragma unroll
        for (int nt = 0; nt < 4; ++nt) {
            const int col = n0 + nt * 16 + l;
            v2f bg; bg.x = pb[col]; bg.y = pb[D1 + col];
            v2f br = ld2(rw + col * D0 + kb);     // res_w1.T: contiguous pair
            accg[nt] = wmma_f32(a1, bg, accg[nt]);
            accr[nt] = wmma_f32(a2, br, accr[nt]);
        }
    }
#pragma unroll
    for (int nt = 0; nt < 4; ++nt) {
        const int col = n0 + nt * 16 + l;
        const float bg = b1[col], br = rb[col];
#pragma unroll
        for (int r = 0; r < 8; ++r) {
            const int row = m0 + r + 8 * half;
            const float xg = accg[nt][r] + bg;
            t1[row * D1 + col] = gelu_exact(xg) + accr[nt][r] + br;
        }
    }
}

// layer2 pre-BN: out = gelu((agg2*dinv_in) @ w2 + b2) + h1   (K = 256)
// Block = 8 waves covering 128 rows x 64 cols. The w2 column panel (64 KB) is
// staged in LDS once per block via CDNA5 async global->LDS copies (ASYNCcnt),
// so each wave's B fragments come from LDS instead of 8x-redundant L2 traffic.
__global__ __launch_bounds__(256) void k_layer2(const float* __restrict__ agg2,
                                                const float* __restrict__ dinv_in,
                                                const float* __restrict__ w2,   // [256,256]
                                                const float* __restrict__ b2,   // [256]
                                                const float* __restrict__ h1,
                                                float* __restrict__ out) {
    __shared__ float bs[D1 * 64];    // w2[:, n0:n0+64], 64 KB

    const int tid = threadIdx.x;
    const int bm = blockIdx.x >> 2;  // 0..255 row-blocks (128 rows each)
    const int bn = blockIdx.x & 3;   // 0..3   col-blocks (64 cols each)
    const int n0 = bn * 64;

    // ---- async stage: 16384 floats = 4096 x b128, 16 per thread ----
    {
        const float* gbase = w2 + n0;
        for (int i = tid; i < 4096; i += 256) {
            const int row = i >> 4;      // 0..255 (k index)
            const int c4  = i & 15;      // float4 within 64-col row
            unsigned long long gaddr =
                (unsigned long long)(const void*)(gbase + row * D1 + c4 * 4);
            // low 32 bits of the generic (flat) LDS pointer = LDS byte address
            unsigned lds_addr = (unsigned)(unsigned long long)(const void*)(bs + i * 4);
            asm volatile("global_load_async_to_lds_b128 %0, %1, off"
                         :: "v"(lds_addr), "v"(gaddr) : "memory");
        }
        asm volatile("s_wait_asynccnt 0x0" ::: "memory");
        __syncthreads();
    }

    const int lane = tid & 31;
    const int wv = tid >> 5;             // 0..7 -> row subtile within block
    const int m0 = bm * 128 + wv * 16;
    const int half = lane >> 4, l = lane & 15;

    const float di = dinv_in[m0 + l];

    v8f acc[4] = {};
    const float* pa = agg2 + (m0 + l) * D1 + 2 * half;

    for (int k0 = 0; k0 < D1; k0 += 4) {
        v2f a = ld2(pa + k0);
        a.x *= di; a.y *= di;
        const int kb = k0 + 2 * half;
        const float* pb = bs + kb * 64;  // LDS
#pragma unroll
        for (int nt = 0; nt < 4; ++nt) {
            const int c = nt * 16 + l;
            v2f b; b.x = pb[c]; b.y = pb[64 + c];
            acc[nt] = wmma_f32(a, b, acc[nt]);
        }
    }
#pragma unroll
    for (int nt = 0; nt < 4; ++nt) {
        const int col = n0 + nt * 16 + l;
        const float bb = b2[col];
#pragma unroll
        for (int r = 0; r < 8; ++r) {
            const int row = m0 + r + 8 * half;
            const float x = acc[nt][r] + bb;
            out[row * D1 + col] = gelu_exact(x) + h1[row * D1 + col];
        }
    }
}

// ---------------- launch ----------------

extern "C" void kernel_launch(void* const* d_in, const int* in_sizes, int n_in,
                              void* d_out, int out_size, void* d_ws, size_t ws_size,
                              hipStream_t stream) {
    const float* nf    = (const float*)d_in[0];   // [N,74]
    const float* Wi    = (const float*)d_in[1];   // [128,74]
    const float* w1    = (const float*)d_in[2];   // [128,256]
    const float* b1    = (const float*)d_in[3];   // [256]
    const float* rw    = (const float*)d_in[4];   // [256,128]
    const float* rb    = (const float*)d_in[5];   // [256]
    const float* bn_g1 = (const float*)d_in[6];
    const float* bn_b1 = (const float*)d_in[7];
    const float* w2    = (const float*)d_in[8];   // [256,256]
    const float* b2    = (const float*)d_in[9];
    const float* bn_g2 = (const float*)d_in[10];
    const float* bn_b2 = (const float*)d_in[11];
    const int*   src   = (const int*)d_in[12];
    const int*   dst   = (const int*)d_in[13];
    float* out = (float*)d_out;

    char* ws = (char*)d_ws;
    float* h0   = (float*)(ws);                       // N*128 (16 MB)
    float* agg1 = (float*)(ws + (size_t)16777216);    // N*128 (16 MB)
    float* t1   = (float*)(ws + (size_t)33554432);    // N*256 (32 MB)
    float* h1   = (float*)(ws + (size_t)67108864);    // N*256 (32 MB)
    float* agg2 = (float*)(ws);                       // N*256, reuses h0+agg1
    float* deg  = (float*)(ws + (size_t)100663296);   // deg_out | deg_in  (2N)
    float* dinv = deg + 2 * N_NODES;                  // dinv_out | dinv_in (2N)
    float* sums = dinv + 2 * N_NODES;                 // sum1|sq1|sum2|sq2 (4*256)

    float* deg_out = deg;
    float* deg_in  = deg + N_NODES;
    float* dinv_out = dinv;
    float* dinv_in  = dinv + N_NODES;

    auto ZG = [](int n) { return (n + 255) / 256; };

    // degrees + stats scratch zero (deg, dinv, sums are contiguous)
    const int small_n = 4 * N_NODES + 4 * D1;
    k_zero<<<ZG(small_n), 256, 0, stream>>>(deg, small_n);
    k_deg<<<ZG(N_EDGES), 256, 0, stream>>>(src, dst, deg_out, deg_in);
    k_dinv<<<ZG(2 * N_NODES), 256, 0, stream>>>(deg, dinv, 2 * N_NODES);

    // h0 = nf @ Wi.T   (4096 waves)
    k_init_gemm<<<512, 256, 0, stream>>>(nf, Wi, h0);

    // aggregation 1 (d=128)
    k_zero<<<ZG(N_NODES * D0), 256, 0, stream>>>(agg1, N_NODES * D0);
    k_scatter<D0><<<ZG(N_EDGES * (D0 / 4)), 256, 0, stream>>>(h0, dinv_out, src, dst, agg1);

    // layer 1 pre-BN (8192 waves)
    k_layer1<<<1024, 256, 0, stream>>>(h0, agg1, dinv_in, w1, b1, rw, rb, t1);
    k_colstats<<<256, 256, 0, stream>>>(t1, sums, sums + 256);
    k_bn<<<ZG(N_NODES * D1), 256, 0, stream>>>(t1, sums, sums + 256, bn_g1, bn_b1, h1);

    // aggregation 2 (d=256) — agg2 aliases h0/agg1 (dead after layer1)
    k_zero<<<ZG(N_NODES * D1), 256, 0, stream>>>(agg2, N_NODES * D1);
    k_scatter<D1><<<ZG(N_EDGES * (D1 / 4)), 256, 0, stream>>>(h1, dinv_out, src, dst, agg2);

    // layer 2 pre-BN straight into d_out (LDS-staged weights), then BN in place
    k_layer2<<<1024, 256, 0, stream>>>(agg2, dinv_in, w2, b2, h1, out);
    k_colstats<<<256, 256, 0, stream>>>(out, sums + 512, sums + 768);
    k_bn<<<ZG(N_NODES * D1), 256, 0, stream>>>(out, sums + 512, sums + 768, bn_g2, bn_b2, out);
}